// QCNN_21818433864145
// MI455X (gfx1250) — compile-verified
//
#include <hip/hip_runtime.h>

typedef __attribute__((ext_vector_type(2))) float v2f;
typedef __attribute__((ext_vector_type(8))) float v8f;

#define NB 4
#define NQ 20
#define NL 5

// ---------------- workspace layout (float offsets) ----------------
#define OFF_URE   ((size_t)0)            // NL*16
#define OFF_UIM   ((size_t)128)          // NL*16
#define OFF_R2RE  ((size_t)256)          // 4*32*32
#define OFF_R2IM  (OFF_R2RE + 4096)
#define OFF_R3RE  (OFF_R2IM + 4096)     // 4*8*8
#define OFF_R3IM  (OFF_R3RE + 256)
#define OFF_R4RE  (OFF_R3IM + 256)      // 4*4*4
#define OFF_R4IM  (OFF_R4RE + 64)
#define OFF_R5RE  (OFF_R4IM + 64)       // 4*2*2
#define OFF_R5IM  (OFF_R5RE + 16)
#define OFF_SRE   ((size_t)16384)                      // 4<<20
#define OFF_SIM   (OFF_SRE + ((size_t)4 << 20))
#define OFF_PRE   (OFF_SIM + ((size_t)4 << 20))
#define OFF_PIM   (OFF_PRE + ((size_t)4 << 20))
// density matrix after pool0 reuses S region
#define OFF_RRE   OFF_SRE
#define OFF_RIM   OFF_SIM

// ---------------- 4x4 complex matmul helper ----------------
__device__ inline void cmm4(const float* ar, const float* ai,
                            const float* br, const float* bi,
                            float* cr, float* ci) {
#pragma unroll
  for (int i = 0; i < 4; ++i)
#pragma unroll
    for (int j = 0; j < 4; ++j) {
      float sr = 0.f, si = 0.f;
#pragma unroll
      for (int k = 0; k < 4; ++k) {
        float xr = ar[i*4+k], xi = ai[i*4+k];
        float yr = br[k*4+j], yi = bi[k*4+j];
        sr += xr*yr - xi*yi;
        si += xr*yi + xi*yr;
      }
      cr[i*4+j] = sr; ci[i*4+j] = si;
    }
}

// U[l] = expm(H[l] - H[l]^H), via scaling-and-squaring Taylor (Horner).
__global__ void build_unitaries(const float* __restrict__ Hre,
                                const float* __restrict__ Him,
                                float* __restrict__ Ure,
                                float* __restrict__ Uim) {
  int l = threadIdx.x;
  if (l >= NL) return;
  float gr[16], gi[16];
#pragma unroll
  for (int i = 0; i < 4; ++i)
#pragma unroll
    for (int j = 0; j < 4; ++j) {
      gr[i*4+j] = Hre[l*16 + i*4+j] - Hre[l*16 + j*4+i];   // Re(A - A^H)
      gi[i*4+j] = Him[l*16 + i*4+j] + Him[l*16 + j*4+i];   // Im(A - A^H)
    }
  // max-row-sum norm estimate -> scale so ||G/2^s|| < 0.25
  float nrm = 0.f;
#pragma unroll
  for (int i = 0; i < 4; ++i) {
    float rs = 0.f;
#pragma unroll
    for (int j = 0; j < 4; ++j)
      rs += sqrtf(gr[i*4+j]*gr[i*4+j] + gi[i*4+j]*gi[i*4+j]);
    nrm = fmaxf(nrm, rs);
  }
  int s = 0;
  while (nrm > 0.25f && s < 40) { nrm *= 0.5f; ++s; }
  float sc = 1.f;
  for (int q = 0; q < s; ++q) sc *= 0.5f;
#pragma unroll
  for (int k = 0; k < 16; ++k) { gr[k] *= sc; gi[k] *= sc; }
  // Horner Taylor: T = I; for n = 14..1: T = I + (G*T)/n
  float tr[16], ti[16], wr[16], wi[16];
#pragma unroll
  for (int k = 0; k < 16; ++k) { tr[k] = (k % 5 == 0) ? 1.f : 0.f; ti[k] = 0.f; }
  for (int n = 14; n >= 1; --n) {
    cmm4(gr, gi, tr, ti, wr, wi);
    float inv = 1.f / (float)n;
#pragma unroll
    for (int k = 0; k < 16; ++k) {
      tr[k] = wr[k]*inv + ((k % 5 == 0) ? 1.f : 0.f);
      ti[k] = wi[k]*inv;
    }
  }
  for (int q = 0; q < s; ++q) {
    cmm4(tr, ti, tr, ti, wr, wi);
#pragma unroll
    for (int k = 0; k < 16; ++k) { tr[k] = wr[k]; ti[k] = wi[k]; }
  }
#pragma unroll
  for (int k = 0; k < 16; ++k) { Ure[l*16+k] = tr[k]; Uim[l*16+k] = ti[k]; }
}

// Generic 2-qubit gate acting on flat-index bit positions (pa, pb) of a
// complex array stored as separate re/im planes. One thread = one group of 4.
__global__ void gate2q(const float* __restrict__ sre, const float* __restrict__ sim,
                       float* __restrict__ dre, float* __restrict__ dim_,
                       const float* __restrict__ Ur, const float* __restrict__ Ui,
                       int conjU, int pa, int pb, unsigned n_groups) {
  unsigned g = blockIdx.x * blockDim.x + threadIdx.x;
  if (g >= n_groups) return;
  int p1 = pa < pb ? pa : pb;
  int p2 = pa < pb ? pb : pa;
  unsigned m1 = (1u << p1) - 1u;
  unsigned x = ((g & ~m1) << 1) | (g & m1);
  unsigned m2 = (1u << p2) - 1u;
  x = ((x & ~m2) << 1) | (x & m2);
  unsigned off[4];
#pragma unroll
  for (int r = 0; r < 2; ++r)
#pragma unroll
    for (int t = 0; t < 2; ++t)
      off[2*r + t] = x + ((unsigned)r << pa) + ((unsigned)t << pb);
  float xr[4], xi[4];
#pragma unroll
  for (int k = 0; k < 4; ++k) { xr[k] = sre[off[k]]; xi[k] = sim[off[k]]; }
  float ur[16], ui[16];
  float sgn = conjU ? -1.f : 1.f;
#pragma unroll
  for (int k = 0; k < 16; ++k) { ur[k] = Ur[k]; ui[k] = sgn * Ui[k]; }
#pragma unroll
  for (int r = 0; r < 4; ++r) {
    float ar = 0.f, ai = 0.f;
#pragma unroll
    for (int k = 0; k < 4; ++k) {
      ar += ur[r*4+k]*xr[k] - ui[r*4+k]*xi[k];
      ai += ur[r*4+k]*xi[k] + ui[r*4+k]*xr[k];
    }
    dre[off[r]] = ar; dim_[off[r]] = ai;
  }
}

// Morton spread: 10 bits -> even bit positions of 20-bit value
__device__ inline unsigned part1by1(unsigned x) {
  x &= 0x0000FFFFu;
  x = (x | (x << 8)) & 0x00FF00FFu;
  x = (x | (x << 4)) & 0x0F0F0F0Fu;
  x = (x | (x << 2)) & 0x33333333u;
  x = (x | (x << 1)) & 0x55555555u;
  return x;
}

// psi flat bit of qubit q is (19-q): traced (even q) bits sit at odd flat
// positions, kept (odd q) bits at even positions. Produce P[b][t][i].
__global__ void deinterleave(const float* __restrict__ sre, const float* __restrict__ sim,
                             float* __restrict__ pre, float* __restrict__ pim) {
  unsigned tid = blockIdx.x * blockDim.x + threadIdx.x;   // < 4<<20
  unsigned b = tid >> 20;
  unsigned t = (tid >> 10) & 1023u;
  unsigned i = tid & 1023u;
  unsigned src = (b << 20) | part1by1(i) | (part1by1(t) << 1);
  pre[tid] = sre[src];
  pim[tid] = sim[src];
}

// Pool 0: rho[b,i,j] = sum_t P[b,t,i] * conj(P[b,t,j])
// Complex 1024x1024x1024 GEMM per batch via V_WMMA_F32_16X16X4_F32.
// rho_re = A^T A + B^T B ; rho_im = B^T A - A^T B  (A=Re P, B=Im P)
__global__ __launch_bounds__(256)
void pool0_gemm(const float* __restrict__ Pre, const float* __restrict__ Pim,
                float* __restrict__ Cre, float* __restrict__ Cim) {
  const int lane = threadIdx.x & 31;
  const int wave = threadIdx.x >> 5;
  const int lo = lane & 15;      // M (for A) / N (for B) index
  const int hi = lane >> 4;      // selects K pair {0,1} vs {2,3}
  const int j0 = blockIdx.x * 16;
  const int i0 = (blockIdx.y * 8 + wave) * 16;
  const int b  = blockIdx.z;
  const size_t pb = (size_t)b << 20;

  v8f accRR = {}, accII = {}, accIR = {}, accRI = {};
  for (int t0 = 0; t0 < 1024; t0 += 4) {
    const size_t r0 = pb + (size_t)(t0 + 2*hi) * 1024;
    const size_t r1 = r0 + 1024;
    v2f ari, aii, arj, aij;
    ari.x = Pre[r0 + i0 + lo]; ari.y = Pre[r1 + i0 + lo];
    aii.x = Pim[r0 + i0 + lo]; aii.y = Pim[r1 + i0 + lo];
    arj.x = Pre[r0 + j0 + lo]; arj.y = Pre[r1 + j0 + lo];
    aij.x = Pim[r0 + j0 + lo]; aij.y = Pim[r1 + j0 + lo];
    accRR = __builtin_amdgcn_wmma_f32_16x16x4_f32(false, ari, false, arj, (short)0, accRR, false, false);
    accII = __builtin_amdgcn_wmma_f32_16x16x4_f32(false, aii, false, aij, (short)0, accII, false, false);
    accIR = __builtin_amdgcn_wmma_f32_16x16x4_f32(false, aii, false, arj, (short)0, accIR, false, false);
    accRI = __builtin_amdgcn_wmma_f32_16x16x4_f32(false, ari, false, aij, (short)0, accRI, false, false);
  }
#pragma unroll
  for (int d = 0; d < 8; ++d) {
    size_t o = pb + (size_t)(i0 + d + 8*hi) * 1024 + j0 + lo;
    Cre[o] = accRR[d] + accII[d];
    Cim[o] = accIR[d] - accRI[d];
  }
}

// Generic density-matrix partial trace over even-indexed qubits.
__global__ void pool_dm(const float* __restrict__ ire, const float* __restrict__ iim,
                        float* __restrict__ ore, float* __restrict__ oim, int nq) {
  const int h = nq / 2;          // #traced
  const int k = nq - h;          // #kept
  unsigned total = (unsigned)NB << (2*k);
  unsigned tid = blockIdx.x * blockDim.x + threadIdx.x;
  if (tid >= total) return;
  unsigned jp = tid & ((1u << k) - 1u);
  unsigned ip = (tid >> k) & ((1u << k) - 1u);
  unsigned b  = tid >> (2*k);
  float sr = 0.f, si = 0.f;
  for (unsigned t = 0; t < (1u << h); ++t) {
    unsigned i = 0, j = 0; int km = 0, tm = 0;
    for (int q = 0; q < nq; ++q) {
      int inbit = nq - 1 - q;
      if ((q % 2 == 0) && (q < 2*h)) {              // traced qubit
        unsigned bt = (t >> (h - 1 - tm)) & 1u;
        i |= bt << inbit; j |= bt << inbit; ++tm;
      } else {                                       // kept qubit
        unsigned bi = (ip >> (k - 1 - km)) & 1u;
        unsigned bj = (jp >> (k - 1 - km)) & 1u;
        i |= bi << inbit; j |= bj << inbit; ++km;
      }
    }
    size_t off = ((((size_t)b << nq) | i) << nq) | j;
    sr += ire[off]; si += iim[off];
  }
  size_t oo = ((((size_t)b << k) | ip) << k) | jp;
  ore[oo] = sr; oim[oo] = si;
}

__global__ void writeout(const float* __restrict__ r5re, float* __restrict__ out) {
  int b = threadIdx.x;
  if (b < NB) out[b] = r5re[b * 4];   // rho0[b,0,0].real
}

// ---------------- host side ----------------
static void get_pairs(int nq, int* pa, int* pb, int* n) {
  int c = 0;
  for (int i = 0; i < nq/2; ++i) { pa[c] = 2*i; pb[c] = 2*i + 1; ++c; }
  if (nq > 2)
    for (int i = 0; i < nq/2; ++i) { pa[c] = 2*i + 1; pb[c] = (2*i + 2) % nq; ++c; }
  *n = c;
}

extern "C" void kernel_launch(void* const* d_in, const int* in_sizes, int n_in,
                              void* d_out, int out_size, void* d_ws, size_t ws_size,
                              hipStream_t stream) {
  (void)in_sizes; (void)n_in; (void)out_size; (void)ws_size;
  const float* psi_re = (const float*)d_in[0];
  const float* psi_im = (const float*)d_in[1];
  const float* H_re   = (const float*)d_in[2];
  const float* H_im   = (const float*)d_in[3];
  float* out = (float*)d_out;
  float* w = (float*)d_ws;

  float* Ure = w + OFF_URE;  float* Uim = w + OFF_UIM;
  float* Sre = w + OFF_SRE;  float* Sim = w + OFF_SIM;
  float* Pre = w + OFF_PRE;  float* Pim = w + OFF_PIM;
  float* Rre = w + OFF_RRE;  float* Rim = w + OFF_RIM;
  float* r2re = w + OFF_R2RE; float* r2im = w + OFF_R2IM;
  float* r3re = w + OFF_R3RE; float* r3im = w + OFF_R3IM;
  float* r4re = w + OFF_R4RE; float* r4im = w + OFF_R4IM;
  float* r5re = w + OFF_R5RE; float* r5im = w + OFF_R5IM;

  // 1) Unitaries for all layers
  build_unitaries<<<1, 32, 0, stream>>>(H_re, H_im, Ure, Uim);

  int pa[32], pb[32], np;

  // 2) Layer 0: gates on the 2^20 state vector (qubit q -> flat bit 19-q)
  get_pairs(NQ, pa, pb, &np);   // 20 pairs
  {
    const unsigned groups = 1u << 20;   // (4<<20)/4
    const dim3 gsz(groups / 256);
    for (int p = 0; p < np; ++p) {
      const float* sr = (p == 0) ? psi_re : Sre;
      const float* si = (p == 0) ? psi_im : Sim;
      gate2q<<<gsz, 256, 0, stream>>>(sr, si, Sre, Sim,
                                      Ure + 0*16, Uim + 0*16, 0,
                                      NQ - 1 - pa[p], NQ - 1 - pb[p], groups);
    }
  }

  // 3) Pool 0: de-interleave into P[b][t][i], then WMMA complex GEMM -> rho
  deinterleave<<<(4u << 20) / 256, 256, 0, stream>>>(Sre, Sim, Pre, Pim);
  pool0_gemm<<<dim3(64, 8, NB), 256, 0, stream>>>(Pre, Pim, Rre, Rim);

  // 4) Layers 1..4: density-matrix gates (ket: flat bit 2nq-1-q, bra: nq-1-q)
  //    followed by partial-trace pools.
  struct { int nq; float *ir, *ii, *orr, *oi; } lay[4] = {
    {10, Rre,  Rim,  r2re, r2im},
    { 5, r2re, r2im, r3re, r3im},
    { 3, r3re, r3im, r4re, r4im},
    { 2, r4re, r4im, r5re, r5im},
  };
  for (int li = 0; li < 4; ++li) {
    const int nq = lay[li].nq;
    const int layer = li + 1;
    get_pairs(nq, pa, pb, &np);
    const unsigned groups = 1u << (2*nq);          // (4<<2nq)/4
    const unsigned blks = (groups + 255) / 256;
    for (int p = 0; p < np; ++p) {
      // ket side: U
      gate2q<<<blks, 256, 0, stream>>>(lay[li].ir, lay[li].ii, lay[li].ir, lay[li].ii,
                                       Ure + layer*16, Uim + layer*16, 0,
                                       2*nq - 1 - pa[p], 2*nq - 1 - pb[p], groups);
      // bra side: conj(U)
      gate2q<<<blks, 256, 0, stream>>>(lay[li].ir, lay[li].ii, lay[li].ir, lay[li].ii,
                                       Ure + layer*16, Uim + layer*16, 1,
                                       nq - 1 - pa[p], nq - 1 - pb[p], groups);
    }
    const int k = nq - nq/2;
    const unsigned tot = (unsigned)NB << (2*k);
    pool_dm<<<(tot + 255) / 256, 256, 0, stream>>>(lay[li].ir, lay[li].ii,
                                                   lay[li].orr, lay[li].oi, nq);
  }

  // 5) P(outcome 0) of the final 1-qubit density matrix
  writeout<<<1, 32, 0, stream>>>(r5re, out);
}